// DiagonalConjugate_8048768713421
// MI455X (gfx1250) — compile-verified
//
#include <hip/hip_runtime.h>

// MI455X / gfx1250 implementation of the diagonal SSM:
//   Y = B@X (WMMA f32, LDS-staged X, fused chunk-local scan)   [kernel 1]
//   cross-chunk carry scan with A^Tc                           [kernel 2]
//   out = Re(C@S_corrected) + D@X (WMMA f32)                   [kernel 3]
// All matrix math uses V_WMMA_F32_16X16X4_F32 (full f32 precision).
// X is staged into LDS transposed (Xt[j][u], row stride 132 dwords) so each
// B-fragment is a single conflict-free ds_load_b64 per WMMA.

typedef __attribute__((ext_vector_type(2))) float v2f;
typedef __attribute__((ext_vector_type(8))) float v8f;

#define T_ 4096
#define N_ 256
#define U_ 128
#define K_ 32
#define P_ 128
#define TC 8            // timesteps per chunk
#define NC (T_ / TC)    // 512 chunks
#define JC (TC * K_)    // 256 columns (j = lt*32 + k) per chunk
#define XSTR 132        // padded LDS row stride (dwords): 132 % 64 == 4

// workspace layout (floats):
//   [0, SZS)          : S_re   (local-then-corrected states, [t][n][k])
//   [SZS, 2*SZS)      : S_im
//   + SZC each        : lastRe, lastIm, prefRe, prefIm   ([chunk][n][k])
#define SZS ((size_t)T_ * N_ * K_)   // 33,554,432
#define SZC ((size_t)NC * N_ * K_)   //  4,194,304

__device__ __forceinline__ v8f wmma4(v2f a, v2f b, v8f c) {
  // D = A(16x4) * B(4x16) + C, f32 — emits v_wmma_f32_16x16x4_f32
  return __builtin_amdgcn_wmma_f32_16x16x4_f32(false, a, false, b, (short)0, c,
                                               false, false);
}

// Stage a quarter-chunk (2 timesteps, 64 j-rows, 8192 floats) of X into LDS,
// transposed to Xt[j][u] with padded stride. xq points at x[t0 + q*2, 0, 0].
__device__ __forceinline__ void stage_x_quarter(const float* __restrict__ xq,
                                                float* __restrict__ XtF,
                                                int tid) {
#pragma unroll
  for (int it = 0; it < 8; ++it) {
    const int e = (it * 256 + tid) * 4;          // float index, 16B aligned
    const float4 v = *(const float4*)(xq + e);   // x[ltl, uu, k..k+3]
    const int k = e & 31;
    const int uu = (e >> 5) & (U_ - 1);
    const int ltl = e >> 12;                     // 0..1
    XtF[(ltl * K_ + k + 0) * XSTR + uu] = v.x;
    XtF[(ltl * K_ + k + 1) * XSTR + uu] = v.y;
    XtF[(ltl * K_ + k + 2) * XSTR + uu] = v.z;
    XtF[(ltl * K_ + k + 3) * XSTR + uu] = v.w;
  }
}

// ---------------------------------------------------------------------------
// Kernel 1: per chunk c and 64-row n-block: Y = B@X via WMMA (X from LDS),
// then chunk-local scan through LDS; writes local states S and last-states.
// 8 waves: wave w -> m-tile (w&3), part (w>>2): 0 = B_re, 1 = B_im.
// ---------------------------------------------------------------------------
__global__ __launch_bounds__(256) void k1_bx_scan(
    const float* __restrict__ x, const float* __restrict__ Are,
    const float* __restrict__ Aim, const float* __restrict__ Bre,
    const float* __restrict__ Bim, float* __restrict__ ws) {
  float* Sre = ws;
  float* Sim = ws + SZS;
  float* lastRe = ws + 2 * SZS;
  float* lastIm = lastRe + SZC;

  __shared__ float smemF[16384];  // 64 KB: Xt quarter (8448 f) / lsY (16384 f)
  float* XtF = smemF;
  float* lsY0 = smemF;            // re-used after GEMM: Y_re [32][JC]
  float* lsY1 = smemF + 32 * JC;  //                     Y_im [32][JC]

  const int tid = threadIdx.x, lane = tid & 31, w = tid >> 5;
  const int mi = w & 3, part = w >> 2;
  const int c = blockIdx.x, nb = blockIdx.y;
  const int n0 = nb * 64 + mi * 16;
  const int t0 = c * TC;
  const float* Bsel = part ? Bim : Bre;
  const float* xc = x + (size_t)t0 * U_ * K_;

  v8f acc[16];
#pragma unroll
  for (int i = 0; i < 16; ++i) acc[i] = v8f{};

  const int arow = n0 + (lane & 15);
  const int ush = (lane >> 4) << 1;  // K offset {0,2} per half-wave
  const int jlane = lane & 15;

  for (int q = 0; q < 4; ++q) {  // quarter-chunks of 2 timesteps
    stage_x_quarter(xc + (size_t)q * 2 * U_ * K_, XtF, tid);
    __syncthreads();
    for (int u0 = 0; u0 < U_; u0 += 4) {
      v2f a = *(const v2f*)(Bsel + (size_t)arow * U_ + u0 + ush);
      const int urow = u0 + ush;
#pragma unroll
      for (int jj = 0; jj < 4; ++jj) {
        const int jl = jj * 16 + jlane;  // local j-row within quarter
        v2f b = *(const v2f*)(&XtF[jl * XSTR + urow]);  // ds_load_b64
        acc[q * 4 + jj] = wmma4(a, b, acc[q * 4 + jj]);
      }
    }
    __syncthreads();
  }

  // Exchange Y through LDS (32 n-rows per phase), then every thread scans
  // 4 independent (n,k) sequences over lt = 0..7.
  for (int phase = 0; phase < 2; ++phase) {
    if ((mi >> 1) == phase) {
      const int nl_base = (mi & 1) * 16 + 8 * (lane >> 4);
      float* dstY = part ? lsY1 : lsY0;
#pragma unroll
      for (int jt = 0; jt < 16; ++jt) {
        const int j = jt * 16 + jlane;
#pragma unroll
        for (int v = 0; v < 8; ++v) dstY[(nl_base + v) * JC + j] = acc[jt][v];
      }
    }
    __syncthreads();
    const int k = tid & 31;
#pragma unroll
    for (int i = 0; i < 4; ++i) {
      const int nl = (tid >> 5) + 8 * i;  // 0..31
      const int ng = nb * 64 + phase * 32 + nl;
      const float ar = Are[ng * K_ + k], ai = Aim[ng * K_ + k];
      float sr = 0.f, si = 0.f;
#pragma unroll
      for (int lt = 0; lt < TC; ++lt) {
        const float yr = lsY0[nl * JC + lt * K_ + k];
        const float yi = lsY1[nl * JC + lt * K_ + k];
        const float nsr = ar * sr - ai * si + yr;
        const float nsi = ar * si + ai * sr + yi;
        sr = nsr;
        si = nsi;
        const size_t o = (size_t)(t0 + lt) * N_ * K_ + (size_t)ng * K_ + k;
        Sre[o] = sr;
        Sim[o] = si;
      }
      const size_t oc = (size_t)c * (N_ * K_) + (size_t)ng * K_ + k;
      lastRe[oc] = sr;
      lastIm[oc] = si;
    }
    __syncthreads();
  }
}

// ---------------------------------------------------------------------------
// Kernel 2: serial carry scan over chunks.  carry[c+1] = A^TC*carry[c]+last[c]
// prefRe/Im[c] = carry *into* chunk c.  8192 independent (n,k) lanes.
// ---------------------------------------------------------------------------
__global__ __launch_bounds__(256) void k2_carry(const float* __restrict__ Are,
                                                const float* __restrict__ Aim,
                                                float* __restrict__ ws) {
  const float* lastRe = ws + 2 * SZS;
  const float* lastIm = lastRe + SZC;
  float* prefRe = ws + 2 * SZS + 2 * SZC;
  float* prefIm = prefRe + SZC;

  const int idx = blockIdx.x * 256 + threadIdx.x;  // == n*K + k
  float pr = Are[idx], pi = Aim[idx];
#pragma unroll
  for (int s = 0; s < 3; ++s) {  // A^8 by repeated squaring
    const float nr = pr * pr - pi * pi, ni = 2.f * pr * pi;
    pr = nr;
    pi = ni;
  }
  float cr = 0.f, ci = 0.f;
  for (int c = 0; c < NC; ++c) {
    const size_t o = (size_t)c * (N_ * K_) + idx;
    prefRe[o] = cr;
    prefIm[o] = ci;
    const float lr = lastRe[o], li = lastIm[o];
    const float nr = pr * cr - pi * ci + lr;
    const float ni = pr * ci + pi * cr + li;
    cr = nr;
    ci = ni;
  }
}

// ---------------------------------------------------------------------------
// Kernel 3: per chunk: out = D@X + Cre@Sre - Cim@Sim, with the carry
// correction  s[lt] += A^(lt+1)*carry  applied while staging S into LDS.
// 8 waves: wave w owns output rows p0 = w*16; inner dim N in 8 slabs of 32.
// ---------------------------------------------------------------------------
__global__ __launch_bounds__(256) void k3_out(
    const float* __restrict__ x, const float* __restrict__ Are,
    const float* __restrict__ Aim, const float* __restrict__ Cre,
    const float* __restrict__ Cim, const float* __restrict__ Dm,
    const float* __restrict__ ws, float* __restrict__ out) {
  const float* Sre = ws;
  const float* Sim = ws + SZS;
  const float* prefRe = ws + 2 * SZS + 2 * SZC;
  const float* prefIm = prefRe + SZC;

  __shared__ float smemF[16384];  // 64 KB: Xt quarter / lsR+lsI slabs
  float* XtF = smemF;
  float* lsR = smemF;             // re-used after D@X: S_re slab [32][JC]
  float* lsI = smemF + 32 * JC;   //                    S_im slab [32][JC]

  const int tid = threadIdx.x, lane = tid & 31, w = tid >> 5;
  const int c = blockIdx.x;
  const int t0 = c * TC;
  const int p0 = w * 16;
  const float* xc = x + (size_t)t0 * U_ * K_;

  v8f acc[16];
#pragma unroll
  for (int i = 0; i < 16; ++i) acc[i] = v8f{};

  const int arow = p0 + (lane & 15);
  const int ush = (lane >> 4) << 1;
  const int jlane = lane & 15;

  // ---- D @ X (inner U = 128), X staged via LDS in quarter-chunks ----
  for (int q = 0; q < 4; ++q) {
    stage_x_quarter(xc + (size_t)q * 2 * U_ * K_, XtF, tid);
    __syncthreads();
    for (int u0 = 0; u0 < U_; u0 += 4) {
      v2f a = *(const v2f*)(Dm + (size_t)arow * U_ + u0 + ush);
      const int urow = u0 + ush;
#pragma unroll
      for (int jj = 0; jj < 4; ++jj) {
        const int jl = jj * 16 + jlane;
        v2f b = *(const v2f*)(&XtF[jl * XSTR + urow]);
        acc[q * 4 + jj] = wmma4(a, b, acc[q * 4 + jj]);
      }
    }
    __syncthreads();
  }

  // ---- Re(C @ S) in 8 slabs of 32 inner rows ----
  for (int s = 0; s < 8; ++s) {
    {  // cooperative corrected slab load: s[lt] += A^(lt+1) * carry_in
      const int k = tid & 31;
#pragma unroll
      for (int i = 0; i < 4; ++i) {
        const int nl = (tid >> 5) + 8 * i;
        const int ng = s * 32 + nl;
        const int idx = ng * K_ + k;
        const float ar = Are[idx], ai = Aim[idx];
        const size_t oc = (size_t)c * (N_ * K_) + idx;
        const float crr = prefRe[oc], cii = prefIm[oc];
        float apr = ar, api = ai;  // A^(lt+1), built incrementally
#pragma unroll
        for (int lt = 0; lt < TC; ++lt) {
          const size_t o = (size_t)(t0 + lt) * N_ * K_ + idx;
          const float sr = Sre[o] + (apr * crr - api * cii);
          const float si = Sim[o] + (apr * cii + api * crr);
          lsR[nl * JC + lt * K_ + k] = sr;
          lsI[nl * JC + lt * K_ + k] = si;
          const float nr = apr * ar - api * ai, ni = apr * ai + api * ar;
          apr = nr;
          api = ni;
        }
      }
    }
    __syncthreads();
    for (int u0 = 0; u0 < 32; u0 += 4) {
      const int ng0 = s * 32 + u0 + ush;
      v2f car = *(const v2f*)(Cre + (size_t)arow * N_ + ng0);
      v2f cai = *(const v2f*)(Cim + (size_t)arow * N_ + ng0);
      cai = -cai;  // out -= Cim@Sim  (f32 WMMA has no A-negate modifier)
      const int ul = u0 + ush;
#pragma unroll
      for (int jt = 0; jt < 16; ++jt) {
        const int j = jt * 16 + jlane;
        v2f br;
        br.x = lsR[ul * JC + j];
        br.y = lsR[(ul + 1) * JC + j];
        acc[jt] = wmma4(car, br, acc[jt]);
        v2f bi;
        bi.x = lsI[ul * JC + j];
        bi.y = lsI[(ul + 1) * JC + j];
        acc[jt] = wmma4(cai, bi, acc[jt]);
      }
    }
    __syncthreads();
  }

  // ---- store out[t, p, k] ----
#pragma unroll
  for (int jt = 0; jt < 16; ++jt) {
    const int lt = jt >> 1;
    const int k = ((jt & 1) << 4) + jlane;
#pragma unroll
    for (int v = 0; v < 8; ++v) {
      const int p = p0 + v + 8 * (lane >> 4);
      out[(size_t)(t0 + lt) * P_ * K_ + (size_t)p * K_ + k] = acc[jt][v];
    }
  }
}

extern "C" void kernel_launch(void* const* d_in, const int* in_sizes, int n_in,
                              void* d_out, int out_size, void* d_ws,
                              size_t ws_size, hipStream_t stream) {
  const float* x = (const float*)d_in[0];
  const float* Are = (const float*)d_in[1];
  const float* Aim = (const float*)d_in[2];
  const float* Bre = (const float*)d_in[3];
  const float* Bim = (const float*)d_in[4];
  const float* Cre = (const float*)d_in[5];
  const float* Cim = (const float*)d_in[6];
  const float* Dm = (const float*)d_in[7];
  float* out = (float*)d_out;
  float* ws = (float*)d_ws;

  k1_bx_scan<<<dim3(NC, N_ / 64), 256, 0, stream>>>(x, Are, Aim, Bre, Bim, ws);
  k2_carry<<<dim3((N_ * K_) / 256), 256, 0, stream>>>(Are, Aim, ws);
  k3_out<<<dim3(NC), 256, 0, stream>>>(x, Are, Aim, Cre, Cim, Dm, ws, out);
}